// Thalamus_67018669686941
// MI455X (gfx1250) — compile-verified
//
#include <hip/hip_runtime.h>
#include <math.h>

// ---------------------------------------------------------------------------
// Thalamus sim (TC/TRN Izhikevich) for MI455X / gfx1250.
// Matvecs run on V_WMMA_F32_16X16X4_F32 (full f32 precision, 16 rows/wave).
// Memory-bound: loop weights (128MB) are L2-resident on the 192MB L2.
// ---------------------------------------------------------------------------

typedef __attribute__((ext_vector_type(2))) float v2f;
typedef __attribute__((ext_vector_type(8))) float v8f;

#define N_NEUR 4096
#define K_EXT  6144

// ---- y = W @ x via WMMA f32 16x16x4. One wave -> 16 output rows. ----------
// A-layout (16x4 f32): lanes 0-15 hold row (lane&15) cols {k,k+1},
//                      lanes 16-31 hold the same rows, cols {k+2,k+3}.
// B is x broadcast across all 16 columns -> every D column equals y;
// we read column 0 (lanes 0 and 16 of the 8 C/D VGPRs).
__global__ __launch_bounds__(128) void matvec_wmma_f32(
    const float* __restrict__ W, const float* __restrict__ x,
    float* __restrict__ y, int K) {
  const int lane = threadIdx.x & 31;
  const int wave = threadIdx.x >> 5;
  const int rb   = (blockIdx.x * 4 + wave) * 16;     // 16 rows per wave
  const int half = lane >> 4;                        // 0: K=k,k+1  1: K=k+2,k+3
  const int kh   = 2 * half;
  const float* wrow = W + (size_t)(rb + (lane & 15)) * (size_t)K;

  v8f c0 = {}, c1 = {}, c2 = {}, c3 = {};
  for (int k = 0; k < K; k += 16) {
    v2f a0 = *(const v2f*)(wrow + k + 0  + kh);
    v2f b0 = *(const v2f*)(x    + k + 0  + kh);
    v2f a1 = *(const v2f*)(wrow + k + 4  + kh);
    v2f b1 = *(const v2f*)(x    + k + 4  + kh);
    v2f a2 = *(const v2f*)(wrow + k + 8  + kh);
    v2f b2 = *(const v2f*)(x    + k + 8  + kh);
    v2f a3 = *(const v2f*)(wrow + k + 12 + kh);
    v2f b3 = *(const v2f*)(x    + k + 12 + kh);
    c0 = __builtin_amdgcn_wmma_f32_16x16x4_f32(false, a0, false, b0, (short)0, c0, false, false);
    c1 = __builtin_amdgcn_wmma_f32_16x16x4_f32(false, a1, false, b1, (short)0, c1, false, false);
    c2 = __builtin_amdgcn_wmma_f32_16x16x4_f32(false, a2, false, b2, (short)0, c2, false, false);
    c3 = __builtin_amdgcn_wmma_f32_16x16x4_f32(false, a3, false, b3, (short)0, c3, false, false);
  }
  v8f c = (c0 + c1) + (c2 + c3);
  // D column 0: lane 0 -> rows rb..rb+7, lane 16 -> rows rb+8..rb+15
  if ((lane & 15) == 0) {
    float* dst = y + rb + 8 * half;
#pragma unroll
    for (int g = 0; g < 8; ++g) dst[g] = c[g];
  }
}

// ---- deterministic sum(|a|), sum(|b|) -> outs[0], outs[1] -----------------
__global__ __launch_bounds__(256) void reduce_abs2(
    const float* __restrict__ a, const float* __restrict__ b,
    float* __restrict__ outs, int n) {
  __shared__ float sa[256];
  __shared__ float sb[256];
  float s0 = 0.f, s1 = 0.f;
  for (int i = threadIdx.x; i < n; i += 256) { s0 += fabsf(a[i]); s1 += fabsf(b[i]); }
  sa[threadIdx.x] = s0; sb[threadIdx.x] = s1;
  __syncthreads();
  for (int off = 128; off > 0; off >>= 1) {
    if ((int)threadIdx.x < off) {
      sa[threadIdx.x] += sa[threadIdx.x + off];
      sb[threadIdx.x] += sb[threadIdx.x + off];
    }
    __syncthreads();
  }
  if (threadIdx.x == 0) { outs[0] = sa[0]; outs[1] = sb[0]; }
}

__device__ __forceinline__ float nd_gain(float sum, float n, float tgt) {
  float m = sum / n + 1e-8f;
  return (m > 1e-3f) ? (tgt / m) : 0.0f;
}

// ---- precompute: normalize I_pal / I_tect, copy state inputs -> ws --------
__global__ void prep_kernel(const float* __restrict__ yp, const float* __restrict__ yt,
                            const float* __restrict__ sums,
                            float* __restrict__ I_pal, float* __restrict__ I_tect,
                            const float* __restrict__ v_tc, const float* __restrict__ u_tc,
                            const float* __restrict__ r_tc, const float* __restrict__ v_trn,
                            const float* __restrict__ u_trn, const float* __restrict__ r_trn,
                            float* __restrict__ st, int n) {
  int i = blockIdx.x * blockDim.x + threadIdx.x;
  if (i >= n) return;
  I_pal[i]  = yp[i] * nd_gain(sums[0], (float)n, 2.0f);
  I_tect[i] = yt[i] * nd_gain(sums[1], (float)n, 6.0f);
  st[0 * n + i] = v_tc[i];
  st[1 * n + i] = u_tc[i];
  st[2 * n + i] = r_tc[i];
  st[3 * n + i] = v_trn[i];
  st[4 * n + i] = u_trn[i];
  st[5 * n + i] = r_trn[i];
}

__device__ __forceinline__ void izh_step(float& v, float& u, float& r, float I,
                                         float a, float b, float c, float d,
                                         float itonic) {
  float It = I + itonic;
  float v1 = v + 0.5f * (0.04f * v * v + 5.0f * v + 140.0f - u + It);
  float v2 = v1 + 0.5f * (0.04f * v1 * v1 + 5.0f * v1 + 140.0f - u + It);
  float u2 = u + a * (b * v - u);
  float spike = (v2 >= 30.0f) ? 1.0f : 0.0f;
  v = (spike > 0.0f) ? c : v2;
  u = u2 + spike * d;
  r = r + (spike - r) * (1.0f / 20.0f);
}

// ---- one substep: normalize drives, Izhikevich update both populations ----
__global__ void update_kernel(const float* __restrict__ y_trn, const float* __restrict__ y_tc,
                              const float* __restrict__ sums,
                              const float* __restrict__ I_pal, const float* __restrict__ I_tect,
                              float* __restrict__ st, int n) {
  int i = blockIdx.x * blockDim.x + threadIdx.x;
  if (i >= n) return;
  float I_trn = y_trn[i] * nd_gain(sums[0], (float)n, 3.0f) + I_pal[i];
  float I_tc  = I_tect[i] - y_tc[i] * nd_gain(sums[1], (float)n, 2.0f) + 1.5f; // na_bias=0.3*5

  float v_trn = st[3 * n + i], u_trn = st[4 * n + i], r_trn = st[5 * n + i];
  izh_step(v_trn, u_trn, r_trn, I_trn, 0.1f, 0.2f, -65.0f, 2.0f, 0.5f);   // FS, I_TONIC_TRN
  float v_tc = st[0 * n + i], u_tc = st[1 * n + i], r_tc = st[2 * n + i];
  izh_step(v_tc, u_tc, r_tc, I_tc, 0.02f, 0.25f, -65.0f, 2.0f, 0.0f);     // LTS, I_TONIC_TC

  st[3 * n + i] = v_trn; st[4 * n + i] = u_trn; st[5 * n + i] = r_trn;
  st[0 * n + i] = v_tc;  st[1 * n + i] = u_tc;  st[2 * n + i] = r_tc;
}

__global__ void finalize_kernel(const float* __restrict__ st, float* __restrict__ out, int n) {
  int i = blockIdx.x * blockDim.x + threadIdx.x;
  if (i >= n) return;
  out[i]     = st[2 * n + i];  // rate_tc
  out[n + i] = st[5 * n + i];  // rate_trn
}

extern "C" void kernel_launch(void* const* d_in, const int* in_sizes, int n_in,
                              void* d_out, int out_size, void* d_ws, size_t ws_size,
                              hipStream_t stream) {
  const float* tect      = (const float*)d_in[0];   // [6144]
  const float* pal       = (const float*)d_in[1];   // [6144]
  const float* W_tect_tc = (const float*)d_in[2];   // [4096 x 6144]
  const float* W_pal_trn = (const float*)d_in[3];   // [4096 x 6144]
  const float* W_tc_trn  = (const float*)d_in[4];   // [4096 x 4096]
  const float* W_trn_tc  = (const float*)d_in[5];   // [4096 x 4096]

  const int N = N_NEUR;
  float* ws     = (float*)d_ws;
  float* yA     = ws;            // raw matvec out (TRN drive / pal precompute)
  float* yB     = ws + N;        // raw matvec out (TC drive / tect precompute)
  float* I_pal  = ws + 2 * N;
  float* I_tect = ws + 3 * N;
  float* st     = ws + 4 * N;    // [v_tc,u_tc,r_tc,v_trn,u_trn,r_trn] x N
  float* sums   = ws + 10 * N;   // [2] reduction scalars

  dim3 mvGrid(N / 64), mvBlk(128);   // 4 waves/block, 16 rows/wave
  dim3 pwGrid(N / 256), pwBlk(256);

  // Precompute substep-invariant drives.
  matvec_wmma_f32<<<mvGrid, mvBlk, 0, stream>>>(W_pal_trn, pal, yA, K_EXT);
  matvec_wmma_f32<<<mvGrid, mvBlk, 0, stream>>>(W_tect_tc, tect, yB, K_EXT);
  reduce_abs2<<<1, 256, 0, stream>>>(yA, yB, sums, N);
  prep_kernel<<<pwGrid, pwBlk, 0, stream>>>(
      yA, yB, sums, I_pal, I_tect,
      (const float*)d_in[6], (const float*)d_in[7], (const float*)d_in[8],
      (const float*)d_in[9], (const float*)d_in[10], (const float*)d_in[11], st, N);

  // 10 serial substeps. Both matvecs use the OLD rates (matches the scan body).
  for (int s = 0; s < 10; ++s) {
    matvec_wmma_f32<<<mvGrid, mvBlk, 0, stream>>>(W_tc_trn, st + 2 * N, yA, N); // W_tc_trn @ r_tc
    matvec_wmma_f32<<<mvGrid, mvBlk, 0, stream>>>(W_trn_tc, st + 5 * N, yB, N); // W_trn_tc @ r_trn
    reduce_abs2<<<1, 256, 0, stream>>>(yA, yB, sums, N);
    update_kernel<<<pwGrid, pwBlk, 0, stream>>>(yA, yB, sums, I_pal, I_tect, st, N);
  }

  finalize_kernel<<<pwGrid, pwBlk, 0, stream>>>(st, (float*)d_out, N);
}